// LSTM_decoder_4647154614633
// MI455X (gfx1250) — compile-verified
//
#include <hip/hip_runtime.h>
#include <math.h>

// ============================================================================
// 2-layer LSTM (B=32, T=1024, IN0=512, H=256), PyTorch gate order (i,f,g,o).
//
//  Phase A: xg0 = X @ Wih0^T + bih0 + bhh0   LDS-tiled, pipelined fp32-WMMA GEMM
//  Phase B: layer-0 recurrence (persistent kernel, in-kernel grid sync)
//  Phase C: xg1 = h1 @ Wih1^T + bih1 + bhh1  same GEMM, K=256
//  Phase D: layer-1 recurrence -> d_out
//
// Workspace (floats): xg [32768*1024] (128 MB) + h1 [32768*256] (32 MB) + 2 ints.
// All matrix math is V_WMMA_F32_16X16X4_F32 (full fp32, matches fp32 reference).
// ============================================================================

typedef __attribute__((ext_vector_type(2))) float v2f;
typedef __attribute__((ext_vector_type(4))) float v4f;
typedef __attribute__((ext_vector_type(8))) float v8f;

#define B_   32
#define T_   1024
#define IN0_ 512
#define H_   256
#define G4H_ 1024  // 4*H

static __device__ __forceinline__ v8f wmma_f32(v2f a, v2f b, v8f c) {
  // D = A(16x4) * B(4x16) + C(16x16), all fp32.
  return __builtin_amdgcn_wmma_f32_16x16x4_f32(
      /*neg_a=*/false, a, /*neg_b=*/false, b,
      /*c_mod=*/(short)0, c, /*reuse_a=*/false, /*reuse_b=*/false);
}

static __device__ __forceinline__ float sigmoidf_(float x) {
  return 1.0f / (1.0f + __expf(-x));
}

// ----------------------------------------------------------------------------
// Phase A/C: Y[M,N] = X[M,K] @ W[N,K]^T + (bA[n] + bB[n]),  N == 1024.
// Block = 8 waves. Block tile: 128 rows x 64 cols. Wave w owns M-tile w.
// B panel [64 cols x 64 k] staged in LDS (double-buffered), shared by all waves.
// K processed in 64-blocks with an even/odd software pipeline: next block's A
// fragments (registers) and B panel (regs -> LDS) are fetched during compute.
// ----------------------------------------------------------------------------
#define BKP 68  // padded LDS k-stride (floats): 272B row -> b128-aligned, no conflicts

__global__ __launch_bounds__(256) void xg_gemm(
    const float* __restrict__ X, const float* __restrict__ W,
    const float* __restrict__ bA, const float* __restrict__ bB,
    float* __restrict__ Y, int M, int N, int K)
{
  __shared__ __attribute__((aligned(16))) float Bl[2][64 * BKP];

  const int tid  = threadIdx.x;
  const int ng   = blockIdx.x & 15;   // N/64 == 16 column groups
  const int bm   = blockIdx.x >> 4;
  const int wv   = tid >> 5;
  const int lane = tid & 31;
  const int half = lane >> 4;
  const int lm   = lane & 15;

  const float* Arow = X + (size_t)(bm * 128 + wv * 16 + lm) * K + 2 * half;

  v4f breg[4];
  auto loadB = [&](int kb) {            // global -> regs (next B panel)
#pragma unroll
    for (int j = 0; j < 4; ++j) {
      const int c = tid + 256 * j;      // 1024 float4 chunks
      const int col = c >> 4, off = (c & 15) * 4;
      breg[j] = *(const v4f*)(W + (size_t)(ng * 64 + col) * K + kb + off);
    }
  };
  auto storeB = [&](int buf) {          // regs -> LDS
#pragma unroll
    for (int j = 0; j < 4; ++j) {
      const int c = tid + 256 * j;
      const int col = c >> 4, off = (c & 15) * 4;
      *(v4f*)(&Bl[buf][col * BKP + off]) = breg[j];
    }
  };
  auto loadA = [&](v2f (&aa)[16], int kb) {
#pragma unroll
    for (int j = 0; j < 16; ++j) aa[j] = *(const v2f*)(Arow + kb + 4 * j);
  };

  v8f acc[4] = {v8f{}, v8f{}, v8f{}, v8f{}};
  auto compute = [&](const v2f (&aa)[16], const float* bb) {
#pragma unroll
    for (int j = 0; j < 16; ++j) {
      v2f b0 = *(const v2f*)(bb + 4 * j);
      v2f b1 = *(const v2f*)(bb + 16 * BKP + 4 * j);
      v2f b2 = *(const v2f*)(bb + 32 * BKP + 4 * j);
      v2f b3 = *(const v2f*)(bb + 48 * BKP + 4 * j);
      acc[0] = wmma_f32(aa[j], b0, acc[0]);
      acc[1] = wmma_f32(aa[j], b1, acc[1]);
      acc[2] = wmma_f32(aa[j], b2, acc[2]);
      acc[3] = wmma_f32(aa[j], b3, acc[3]);
    }
  };

  v2f a0[16], a1[16];
  const float* bb0 = &Bl[0][lm * BKP + 2 * half];
  const float* bb1 = &Bl[1][lm * BKP + 2 * half];

  loadB(0);
  loadA(a0, 0);
  storeB(0);
  __syncthreads();

  const int nblk = K >> 6;  // 8 (K=512) or 4 (K=256); always even
  for (int blk = 0; blk < nblk; blk += 2) {
    // even phase: compute (a0, buf0); prefetch blk+1 into (a1, buf1)
    loadB((blk + 1) << 6);
    loadA(a1, (blk + 1) << 6);
    compute(a0, bb0);
    storeB(1);
    __syncthreads();
    // odd phase: compute (a1, buf1); prefetch blk+2 into (a0, buf0)
    if (blk + 2 < nblk) {
      loadB((blk + 2) << 6);
      loadA(a0, (blk + 2) << 6);
    }
    compute(a1, bb1);
    if (blk + 2 < nblk) storeB(0);
    __syncthreads();
  }

  const int nc = ng * 64 + lm;
  const float bi0 = bA[nc]      + bB[nc];
  const float bi1 = bA[nc + 16] + bB[nc + 16];
  const float bi2 = bA[nc + 32] + bB[nc + 32];
  const float bi3 = bA[nc + 48] + bB[nc + 48];
#pragma unroll
  for (int r = 0; r < 8; ++r) {
    const size_t row = (size_t)(bm * 128 + wv * 16 + 8 * half + r) * N;
    Y[row + nc]      = acc[0][r] + bi0;
    Y[row + nc + 16] = acc[1][r] + bi1;
    Y[row + nc + 32] = acc[2][r] + bi2;
    Y[row + nc + 48] = acc[3][r] + bi3;
  }
}

// ----------------------------------------------------------------------------
// Phase B/D: persistent LSTM recurrence. 16 WGs x 256 threads. WG `wg` owns
// hidden columns [wg*16, wg*16+16) of all 4 gate blocks (64 gate columns).
// Per step: stage h_{t-1} [32x256] into LDS with coalesced b128 loads, then
// 8 waves each run one 16x16 WMMA tile fed entirely from LDS, gates mix via
// LDS, h written to global, grid-wide sync (atomic counter + agent fences).
// ----------------------------------------------------------------------------
#define HP 260  // padded LDS k-stride (floats): 1040B row -> b128-aligned, no conflicts

__global__ __launch_bounds__(256) void lstm_rec(
    const float* __restrict__ xg,    // [B*T, 4H], row index = b*T + t
    const float* __restrict__ Whh,   // [4H, H] row-major
    float* __restrict__ hout,        // [B, T, H]
    int* __restrict__ counter, int nwg)
{
  extern __shared__ __attribute__((aligned(16))) float smem[];
  float* Wl = smem;             // [64][HP]  weights (col = g*16+n, k-major)
  float* hl = Wl + 64 * HP;     // [32][HP]  staged h_{t-1}
  float* gl = hl + 32 * HP;     // [32][64]  gate staging: b*64 + g*16 + n
  float* cl = gl + 32 * 64;     // [32][16]  cell state: b*16 + n

  const int wg  = blockIdx.x;
  const int n0  = wg * 16;
  const int tid = threadIdx.x;

  // Stage this WG's 64 Whh rows (4 gates x 16 hidden) into padded LDS (b128).
#pragma unroll
  for (int j = 0; j < 16; ++j) {
    const int c = tid + 256 * j;        // 4096 float4 chunks
    const int col = c >> 6, off = (c & 63) * 4;
    const int g = col >> 4, n = col & 15;
    v4f w = *(const v4f*)(Whh + (size_t)(g * 256 + n0 + n) * H_ + off);
    *(v4f*)(Wl + col * HP + off) = w;
  }
  for (int i = tid; i < 32 * 16; i += 256) cl[i] = 0.0f;  // c_0 = 0
  __syncthreads();

  const int wv   = tid >> 5;
  const int lane = tid & 31;
  const int half = lane >> 4;
  const int lm   = lane & 15;
  const int mt   = wv & 1;        // batch tile (rows 16*mt .. +16)
  const int g    = wv >> 1;       // gate block 0..3
  const float* ab = hl + (mt * 16 + lm) * HP + 2 * half;  // A frag base (LDS)
  const float* wb = Wl + (g * 16 + lm) * HP + 2 * half;   // B frag base (LDS)
  const int xcol = g * 256 + n0 + lm;                     // xg column

  for (int t = 0; t < T_; ++t) {
    // C init = input projection (biases folded in at phase A/C).
    v8f acc = {};
#pragma unroll
    for (int r = 0; r < 8; ++r) {
      const int b = mt * 16 + 8 * half + r;
      acc[r] = xg[((size_t)b * T_ + t) * G4H_ + xcol];
      if (t + 1 < T_)  // warm L2 for next step's input projection
        __builtin_prefetch(&xg[((size_t)b * T_ + t + 1) * G4H_ + xcol], 0, 1);
    }

    if (t > 0) {  // cooperatively stage h_{t-1} into LDS (coalesced b128)
#pragma unroll
      for (int j = 0; j < 8; ++j) {
        const int c = tid + 256 * j;    // 2048 float4 chunks
        const int row = c >> 6, off = (c & 63) * 4;
        v4f hv = *(const v4f*)(hout + ((size_t)row * T_ + (t - 1)) * H_ + off);
        *(v4f*)(hl + row * HP + off) = hv;
      }
    }
    __syncthreads();

    if (t > 0) {  // h_{-1} = 0 -> skip GEMM at t==0
#pragma unroll 8
      for (int k = 0; k < H_; k += 4) {
        v2f a = *(const v2f*)(ab + k);
        v2f b = *(const v2f*)(wb + k);
        acc = wmma_f32(a, b, acc);
      }
    }

    // Stage gate tile to LDS for cross-gate mixing.
#pragma unroll
    for (int r = 0; r < 8; ++r) {
      const int b = mt * 16 + 8 * half + r;
      gl[b * 64 + g * 16 + lm] = acc[r];
    }
    __syncthreads();

    // Elementwise cell update: 512 (b,n) pairs, 2 per thread.
#pragma unroll
    for (int e = 0; e < 2; ++e) {
      const int idx = tid * 2 + e;
      const int b = idx >> 4, n = idx & 15;
      const float i_ = sigmoidf_(gl[b * 64 +      n]);
      const float f_ = sigmoidf_(gl[b * 64 + 16 + n]);
      const float g_ = tanhf    (gl[b * 64 + 32 + n]);
      const float o_ = sigmoidf_(gl[b * 64 + 48 + n]);
      const float c  = f_ * cl[idx] + i_ * g_;
      cl[idx] = c;
      hout[((size_t)b * T_ + t) * H_ + n0 + n] = o_ * tanhf(c);
    }

    // Grid-wide sync: release h writes, arrive, spin, acquire.
    __builtin_amdgcn_fence(__ATOMIC_RELEASE, "agent");
    __syncthreads();
    if (tid == 0) {
      atomicAdd(counter, 1);
      const int target = nwg * (t + 1);
      while (__hip_atomic_load(counter, __ATOMIC_RELAXED,
                               __HIP_MEMORY_SCOPE_AGENT) < target) {
        __builtin_amdgcn_s_sleep(2);
      }
    }
    __syncthreads();
    __builtin_amdgcn_fence(__ATOMIC_ACQUIRE, "agent");  // invalidate WGP$
  }
}

__global__ void init_counters(int* c) {
  if (threadIdx.x < 2) c[threadIdx.x] = 0;
}

// ----------------------------------------------------------------------------
extern "C" void kernel_launch(void* const* d_in, const int* in_sizes, int n_in,
                              void* d_out, int out_size, void* d_ws,
                              size_t ws_size, hipStream_t stream) {
  const float* x    = (const float*)d_in[0];
  const float* Wih0 = (const float*)d_in[1];
  const float* Whh0 = (const float*)d_in[2];
  const float* bih0 = (const float*)d_in[3];
  const float* bhh0 = (const float*)d_in[4];
  const float* Wih1 = (const float*)d_in[5];
  const float* Whh1 = (const float*)d_in[6];
  const float* bih1 = (const float*)d_in[7];
  const float* bhh1 = (const float*)d_in[8];
  float* out = (float*)d_out;

  float* xg = (float*)d_ws;                       // [32768, 1024]
  float* h1 = xg + (size_t)B_ * T_ * G4H_;        // [32768, 256]
  int* counters = (int*)(h1 + (size_t)B_ * T_ * H_);

  const size_t recLds =
      (size_t)(64 * HP + 32 * HP + 32 * 64 + 32 * 16) * sizeof(float);
  const int M = B_ * T_;
  const int gemmBlocks = (M / 128) * (G4H_ / 64);  // 256 * 16 = 4096

  init_counters<<<1, 32, 0, stream>>>(counters);

  // Phase A: xg0 = X @ Wih0^T + bih0 + bhh0   (K = 512)
  xg_gemm<<<gemmBlocks, 256, 0, stream>>>(x, Wih0, bih0, bhh0, xg,
                                          M, G4H_, IN0_);
  // Phase B: layer-0 recurrence -> h1
  lstm_rec<<<16, 256, recLds, stream>>>(xg, Whh0, h1, counters + 0, 16);

  // Phase C: xg1 = h1 @ Wih1^T + bih1 + bhh1  (K = 256)
  xg_gemm<<<gemmBlocks, 256, 0, stream>>>(h1, Wih1, bih1, bhh1, xg,
                                          M, G4H_, H_);
  // Phase D: layer-1 recurrence -> d_out
  lstm_rec<<<16, 256, recLds, stream>>>(xg, Whh1, out, counters + 1, 16);
}